// Net_6227702579590
// MI455X (gfx1250) — compile-verified
//
#include <hip/hip_runtime.h>

typedef __attribute__((ext_vector_type(2))) float v2f;
typedef __attribute__((ext_vector_type(8))) float v8f;

#define ALPHA    0.1f
#define KPROP    10
#define NNODES   100000
#define NEDGES   3200000
#define NFEAT    512
#define NHID     256
#define NCLS     64

// ---------------- graph normalization ----------------

__global__ __launch_bounds__(256) void deg_kernel(const int* __restrict__ col,
                                                  unsigned* __restrict__ cnt, int nE) {
    int e = blockIdx.x * 256 + threadIdx.x;
    if (e < nE) atomicAdd(&cnt[col[e]], 1u);
}

__global__ __launch_bounds__(256) void dinv_kernel(const unsigned* __restrict__ cnt,
                                                   float* __restrict__ dinv,
                                                   float* __restrict__ selfnorm, int n) {
    int i = blockIdx.x * 256 + threadIdx.x;
    if (i < n) {
        float d = rsqrtf((float)(cnt[i] + 1u));   // +1 = self loop
        dinv[i] = d;
        selfnorm[i] = d * d;
    }
}

__global__ __launch_bounds__(256) void norm_kernel(const int* __restrict__ ei,
                                                   const float* __restrict__ dinv,
                                                   float* __restrict__ norm, int nE) {
    int e = blockIdx.x * 256 + threadIdx.x;
    if (e < nE) {
        int r = ei[e];            // row = edge_index[0]
        int c = ei[nE + e];       // col = edge_index[1]
        norm[e] = dinv[r] * dinv[c];
    }
}

// ---------------- WMMA fp32 GEMM: C = act(A[MxK] * B[KxN] + bias) ----------------
// Each wave computes a 16x64 output tile using V_WMMA_F32_16X16X4_F32.
// A layout (16x4 f32): lanes 0-15 -> M=lane, v0=K0 v1=K1; lanes 16-31 -> M=lane-16, v0=K2 v1=K3.
// B layout (4x16 f32): lanes 0-15 -> N=lane, v0=K0 v1=K1; lanes 16-31 -> N=lane-16, v0=K2 v1=K3.
// C/D layout (16x16 f32): vgpr r -> row r (lanes 0-15, N=lane) / row r+8 (lanes 16-31, N=lane-16).

template <bool RELU>
__global__ __launch_bounds__(256) void gemm_wmma(const float* __restrict__ A,
                                                 const float* __restrict__ B,
                                                 const float* __restrict__ bias,
                                                 float* __restrict__ C,
                                                 int M, int K, int N) {
    const int lane  = threadIdx.x & 31;
    const int wave  = (blockIdx.x * 256 + threadIdx.x) >> 5;
    const int tilesN = N >> 6;
    const int tileM = wave / tilesN;
    const int tileN = wave % tilesN;
    if (tileM * 16 >= M) return;

    const int l15 = lane & 15;
    const int hi  = lane >> 4;

    const float* aRow = A + (size_t)(tileM * 16 + l15) * K + hi * 2;
    const int n0 = tileN * 64 + l15;

    v8f acc0 = {}, acc1 = {}, acc2 = {}, acc3 = {};

    for (int k = 0; k < K; k += 4) {
        v2f a = *(const v2f*)(aRow + k);
        const float* bp = B + (size_t)(k + hi * 2) * N + n0;
        v2f b0 = { bp[0],  bp[N] };
        v2f b1 = { bp[16], bp[N + 16] };
        v2f b2 = { bp[32], bp[N + 32] };
        v2f b3 = { bp[48], bp[N + 48] };
        acc0 = __builtin_amdgcn_wmma_f32_16x16x4_f32(false, a, false, b0, (short)0, acc0, false, false);
        acc1 = __builtin_amdgcn_wmma_f32_16x16x4_f32(false, a, false, b1, (short)0, acc1, false, false);
        acc2 = __builtin_amdgcn_wmma_f32_16x16x4_f32(false, a, false, b2, (short)0, acc2, false, false);
        acc3 = __builtin_amdgcn_wmma_f32_16x16x4_f32(false, a, false, b3, (short)0, acc3, false, false);
    }

    const int mBase = tileM * 16 + hi * 8;
#define EPILOG(ACC, T)                                                              \
    {                                                                               \
        const int n = tileN * 64 + (T) * 16 + l15;                                  \
        const float bia = bias[n];                                                  \
        _Pragma("unroll")                                                           \
        for (int r = 0; r < 8; ++r) {                                               \
            float v = ACC[r] + bia;                                                 \
            if (RELU) v = fmaxf(v, 0.0f);                                           \
            C[(size_t)(mBase + r) * N + n] = v;                                     \
        }                                                                           \
    }
    EPILOG(acc0, 0) EPILOG(acc1, 1) EPILOG(acc2, 2) EPILOG(acc3, 3)
#undef EPILOG
}

// ---------------- propagation ----------------

// agg = ALPHA*h + (1-ALPHA)*selfnorm[node]*cur   (self-loop term + teleport term)
__global__ __launch_bounds__(256) void prop_init(const float* __restrict__ h,
                                                 const float* __restrict__ cur,
                                                 const float* __restrict__ selfnorm,
                                                 float* __restrict__ agg, int total) {
    int i = blockIdx.x * 256 + threadIdx.x;
    if (i < total)
        agg[i] = ALPHA * h[i] + (1.0f - ALPHA) * selfnorm[i >> 6] * cur[i];
}

// one wave per edge: lane handles classes (2*lane, 2*lane+1); 256B coalesced gather,
// fp32 atomic scatter-add into the destination row (L2 atomic units).
__global__ __launch_bounds__(256) void scatter_kernel(const int* __restrict__ ei,
                                                      const float* __restrict__ norm,
                                                      const float* __restrict__ cur,
                                                      float* __restrict__ agg, int nE) {
    int gid  = blockIdx.x * 256 + threadIdx.x;
    int e    = gid >> 5;
    int lane = gid & 31;
    if (e >= nE) return;
    int r = ei[e];          // destination (segment_sum over row)
    int c = ei[nE + e];     // source (gather out[col])
    float w = norm[e] * (1.0f - ALPHA);
    const float2* src = (const float2*)(cur + (size_t)c * NCLS);
    float2 v = src[lane];
    float* dst = agg + (size_t)r * NCLS + 2 * lane;
    atomicAdd(dst,     w * v.x);
    atomicAdd(dst + 1, w * v.y);
}

// ---------------- log-softmax (wave per node, 64 classes = 2 per lane) ----------------

__global__ __launch_bounds__(256) void logsoftmax_kernel(const float* __restrict__ cur,
                                                         float* __restrict__ out, int n) {
    int gid  = blockIdx.x * 256 + threadIdx.x;
    int node = gid >> 5;
    int lane = gid & 31;
    if (node >= n) return;
    const float2* p = (const float2*)(cur + (size_t)node * NCLS);
    float2 v = p[lane];
    float m = fmaxf(v.x, v.y);
    #pragma unroll
    for (int off = 16; off > 0; off >>= 1) m = fmaxf(m, __shfl_xor(m, off, 32));
    float s = __expf(v.x - m) + __expf(v.y - m);
    #pragma unroll
    for (int off = 16; off > 0; off >>= 1) s += __shfl_xor(s, off, 32);
    float l = __logf(s) + m;
    float2 res; res.x = v.x - l; res.y = v.y - l;
    ((float2*)(out + (size_t)node * NCLS))[lane] = res;
}

// ---------------- host driver ----------------

extern "C" void kernel_launch(void* const* d_in, const int* in_sizes, int n_in,
                              void* d_out, int out_size, void* d_ws, size_t ws_size,
                              hipStream_t stream) {
    const float* x  = (const float*)d_in[0];
    const int*   ei = (const int*)d_in[1];     // [2, NEDGES] row-major (row ; col)
    const float* W1 = (const float*)d_in[2];
    const float* b1 = (const float*)d_in[3];
    const float* W2 = (const float*)d_in[4];
    const float* b2 = (const float*)d_in[5];
    float* out = (float*)d_out;

    char* ws = (char*)d_ws;
    size_t off = 0;
    auto carve = [&](size_t bytes) -> void* {
        void* p = ws + off;
        off = (off + bytes + 255) & ~(size_t)255;
        return p;
    };
    unsigned* cnt      = (unsigned*)carve((size_t)NNODES * 4);
    float*    dinv     = (float*)carve((size_t)NNODES * 4);
    float*    selfnorm = (float*)carve((size_t)NNODES * 4);
    float*    norm     = (float*)carve((size_t)NEDGES * 4);
    float*    h        = (float*)carve((size_t)NNODES * NCLS * 4);
    char*     region   = (char*)carve((size_t)NNODES * NHID * 4);  // h1, then ping-pong bufs
    float* h1   = (float*)region;
    float* bufA = (float*)region;
    float* bufB = (float*)(region + (size_t)NNODES * NCLS * 4);

    // 1. degrees / normalization
    hipMemsetAsync(cnt, 0, (size_t)NNODES * 4, stream);
    deg_kernel<<<(NEDGES + 255) / 256, 256, 0, stream>>>(ei + NEDGES, cnt, NEDGES);
    dinv_kernel<<<(NNODES + 255) / 256, 256, 0, stream>>>(cnt, dinv, selfnorm, NNODES);
    norm_kernel<<<(NEDGES + 255) / 256, 256, 0, stream>>>(ei, dinv, norm, NEDGES);

    // 2. MLP via fp32 WMMA
    int waves1 = (NNODES / 16) * (NHID / 64);   // 25000 waves
    gemm_wmma<true ><<<(waves1 * 32 + 255) / 256, 256, 0, stream>>>(x,  W1, b1, h1, NNODES, NFEAT, NHID);
    int waves2 = (NNODES / 16) * (NCLS / 64);   // 6250 waves
    gemm_wmma<false><<<(waves2 * 32 + 255) / 256, 256, 0, stream>>>(h1, W2, b2, h,  NNODES, NHID, NCLS);

    // 3. K propagation steps (ping-pong; working set stays L2-resident on MI455X)
    const int total = NNODES * NCLS;
    const float* cur = h;
    for (int it = 0; it < KPROP; ++it) {
        float* agg = (it & 1) ? bufB : bufA;
        prop_init<<<(total + 255) / 256, 256, 0, stream>>>(h, cur, selfnorm, agg, total);
        scatter_kernel<<<((size_t)NEDGES * 32 + 255) / 256, 256, 0, stream>>>(ei, norm, cur, agg, NEDGES);
        cur = agg;
    }

    // 4. log-softmax
    logsoftmax_kernel<<<((size_t)NNODES * 32 + 255) / 256, 256, 0, stream>>>(cur, out, NNODES);
}